// AFGNet_25228637897523
// MI455X (gfx1250) — compile-verified
//
#include <hip/hip_runtime.h>
#include <math.h>

#define F_DIM    64
#define N_NODES  2048
#define N_HEADS  4
#define HEAD_DIM 16
#define C_DIM    8
#define KA       24            // augmented K dim: 16 (head) + 8 (low-rank bias)
#define BATCH    4
#define TW       4             // column tiles per softmax pass
#define CW       (TW * 16)     // 64 columns per outer iteration
#define NOUT     (N_NODES / CW)

typedef __attribute__((ext_vector_type(2))) float v2f;
typedef __attribute__((ext_vector_type(8))) float v8f;

// ---- 16-lane butterfly reductions in the VALU pipe (DPP16, no LDS) --------
template <int CTRL>
__device__ __forceinline__ float dpp_bfly(float x) {
    int xi = __float_as_int(x);
    return __int_as_float(
        __builtin_amdgcn_update_dpp(xi, xi, CTRL, 0xF, 0xF, true));
}
__device__ __forceinline__ float red16_max(float x) {
    x = fmaxf(x, dpp_bfly<0xB1>(x));   // quad_perm [1,0,3,2]  (xor 1)
    x = fmaxf(x, dpp_bfly<0x4E>(x));   // quad_perm [2,3,0,1]  (xor 2)
    x = fmaxf(x, dpp_bfly<0x141>(x));  // row_half_mirror
    x = fmaxf(x, dpp_bfly<0x140>(x));  // row_mirror
    return x;
}
__device__ __forceinline__ float red16_sum(float x) {
    x += dpp_bfly<0xB1>(x);
    x += dpp_bfly<0x4E>(x);
    x += dpp_bfly<0x141>(x);
    x += dpp_bfly<0x140>(x);
    return x;
}

// LDS byte offset of a __shared__ object (generic LDS pointer low 32 bits)
__device__ __forceinline__ uint32_t lds_off(const void* p) {
    return (uint32_t)(uintptr_t)p;
}

// Issue one thread's share of an async tile fetch: 8 floats of `learned`
// and 8 floats of `fixed` straight into LDS (2x b128 each, ASYNCcnt-tracked).
__device__ __forceinline__ void issue_tile_async(const float* gl, const float* gf,
                                                 uint32_t ldst_l, uint32_t ldst_f) {
    asm volatile(
        "global_load_async_to_lds_b128 %0, %2, off\n\t"
        "global_load_async_to_lds_b128 %0, %2, off offset:16\n\t"
        "global_load_async_to_lds_b128 %1, %3, off\n\t"
        "global_load_async_to_lds_b128 %1, %3, off offset:16"
        :
        : "v"(ldst_l), "v"(ldst_f), "v"(gl), "v"(gf)
        : "memory");
}

// ---------------------------------------------------------------------------
// Kernel 1: qkv = (x@Wqc+bqc)@Wqe+bqe ; build augmented qa/ka and transposed vT
// ---------------------------------------------------------------------------
__global__ __launch_bounds__(256) void prep_kernel(
    const float* __restrict__ feat, const float* __restrict__ Wqc,
    const float* __restrict__ bqc,  const float* __restrict__ Wqe,
    const float* __restrict__ bqe,  const float* __restrict__ bf1,
    const float* __restrict__ bf2,
    float* __restrict__ qa, float* __restrict__ ka, float* __restrict__ vT)
{
    __shared__ float sWqc[F_DIM * 3 * C_DIM];
    __shared__ float sWqe[3 * C_DIM * 3 * F_DIM];
    __shared__ float sbqc[3 * C_DIM];
    __shared__ float sbqe[3 * F_DIM];

    for (int i = threadIdx.x; i < F_DIM * 3 * C_DIM; i += 256) sWqc[i] = Wqc[i];
    for (int i = threadIdx.x; i < 3 * C_DIM * 3 * F_DIM; i += 256) sWqe[i] = Wqe[i];
    if (threadIdx.x < 24) sbqc[threadIdx.x] = bqc[threadIdx.x];
    for (int i = threadIdx.x; i < 192; i += 256) sbqe[i] = bqe[i];
    __syncthreads();

    const int row = blockIdx.x * 256 + threadIdx.x;
    const int b = row >> 11;
    const int n = row & (N_NODES - 1);

    float x[F_DIM];
    #pragma unroll
    for (int d = 0; d < F_DIM; ++d) x[d] = feat[(size_t)row * F_DIM + d];

    float t[24];
    #pragma unroll
    for (int j = 0; j < 24; ++j) {
        float acc = sbqc[j];
        #pragma unroll
        for (int d = 0; d < F_DIM; ++d) acc += x[d] * sWqc[d * 24 + j];
        t[j] = acc;
    }

    for (int j = 0; j < 192; ++j) {
        float y = sbqe[j];
        #pragma unroll
        for (int c = 0; c < 24; ++c) y += t[c] * sWqe[c * 192 + j];
        if (j < 64) {
            int h = j >> 4, d = j & 15;
            qa[(((size_t)b * N_HEADS + h) * N_NODES + n) * KA + d] = y * 0.25f;
        } else if (j < 128) {
            int jj = j - 64, h = jj >> 4, d = jj & 15;
            ka[(((size_t)b * N_HEADS + h) * N_NODES + n) * KA + d] = y;
        } else {
            int jj = j - 128, h = jj >> 4, d = jj & 15;
            vT[(((size_t)b * N_HEADS + h) * HEAD_DIM + d) * N_NODES + n] = y;
        }
    }
    #pragma unroll
    for (int h = 0; h < N_HEADS; ++h)
        #pragma unroll
        for (int c = 0; c < C_DIM; ++c) {
            qa[(((size_t)b * N_HEADS + h) * N_NODES + n) * KA + 16 + c] =
                bf1[((size_t)h * N_NODES + n) * C_DIM + c];
            ka[(((size_t)b * N_HEADS + h) * N_NODES + n) * KA + 16 + c] =
                bf2[((size_t)h * C_DIM + c) * N_NODES + n];
        }
}

// ---------------------------------------------------------------------------
// Kernel 2: fused flash attention. 1 workgroup = (b, 16-row block); 4 waves =
// 4 heads. The graph tiles (the HBM-bound 134MB stream) are double-buffered in
// LDS via global_load_async_to_lds_b128 one iteration ahead (ASYNCcnt).
// Per pass: 64 columns = 4x (6 fp32 WMMAs), DPP-butterfly online softmax,
// 4x (4 fp32 WMMAs) for P@V.
// ---------------------------------------------------------------------------
__global__ __launch_bounds__(128) void attn_kernel(
    const float* __restrict__ qa, const float* __restrict__ ka,
    const float* __restrict__ vT, const float* __restrict__ learned,
    const float* __restrict__ fixedg, const float* __restrict__ gfusion,
    float* __restrict__ ctx)
{
    __shared__ float lds_l[2][16 * CW];        // raw learned tiles (8 KB)
    __shared__ float lds_f[2][16 * CW];        // raw fixed tiles   (8 KB)
    __shared__ float lds_p[N_HEADS][TW * 256]; // P layout-shuffle  (16 KB)

    const int h    = threadIdx.x >> 5;
    const int lane = threadIdx.x & 31;
    const int lg   = lane >> 4;
    const int lm   = lane & 15;
    const int b    = blockIdx.x >> 7;
    const int n0   = (blockIdx.x & (N_NODES / 16 - 1)) << 4;

    const float gf = gfusion[0];
    const float fr = 1.0f / (1.0f + __expf(-gf));
    const float fi = 1.0f - fr;

    v2f aq[6];
    const float* qbase =
        qa + (((size_t)b * N_HEADS + h) * N_NODES + n0 + lm) * KA + lg * 2;
    #pragma unroll
    for (int s = 0; s < 6; ++s) aq[s] = *(const v2f*)(qbase + s * 4);

    v8f O = {0.f, 0.f, 0.f, 0.f, 0.f, 0.f, 0.f, 0.f};
    float mrow[8], lrow[8];
    #pragma unroll
    for (int r = 0; r < 8; ++r) { mrow[r] = -3.0e38f; lrow[r] = 0.0f; }

    const float* kb0 = ka + ((size_t)b * N_HEADS + h) * N_NODES * KA;
    const float* vb0 = vT + (((size_t)b * N_HEADS + h) * HEAD_DIM + lm) * N_NODES;
    float* pw = &lds_p[h][0];

    // per-thread slice of the async tile fetch: row rr, 8 cols starting at cc
    const int rr = threadIdx.x >> 3;
    const int cc = (threadIdx.x & 7) << 3;
    const float* gl = learned + ((size_t)b * N_NODES + n0 + rr) * N_NODES + cc;
    const float* gfx = fixedg + (size_t)(n0 + rr) * N_NODES + cc;
    uint32_t dl[2], df[2];
    #pragma unroll
    for (int u = 0; u < 2; ++u) {
        dl[u] = lds_off(&lds_l[u][rr * CW + cc]);
        df[u] = lds_off(&lds_f[u][rr * CW + cc]);
    }

    // prologue: tile 0 in flight
    issue_tile_async(gl, gfx, dl[0], df[0]);
    gl += CW; gfx += CW;

    for (int j = 0; j < NOUT; ++j) {
        const int c0 = j * CW;

        // ---- prefetch tile j+1 while we work on tile j --------------------
        if (j + 1 < NOUT) {
            issue_tile_async(gl, gfx, dl[(j + 1) & 1], df[(j + 1) & 1]);
            gl += CW; gfx += CW;
            asm volatile("s_wait_asynccnt 0x4" ::: "memory"); // tile j landed
        } else {
            asm volatile("s_wait_asynccnt 0x0" ::: "memory");
        }
        __syncthreads();   // tile j visible to all 4 head-waves

        const float* cl = &lds_l[j & 1][0];
        const float* cf = &lds_f[j & 1][0];

        // ---- S[tt] = [q/4 | bf1] @ [k^T ; bf2] + fr*L + (1-fr)*F ----------
        v8f S[TW];
        #pragma unroll
        for (int tt = 0; tt < TW; ++tt) {
            v8f acc = {0.f, 0.f, 0.f, 0.f, 0.f, 0.f, 0.f, 0.f};
            const float* kbase = kb0 + (size_t)(c0 + tt * 16 + lm) * KA + lg * 2;
            #pragma unroll
            for (int s = 0; s < 6; ++s) {
                v2f bk = *(const v2f*)(kbase + s * 4);
                acc = __builtin_amdgcn_wmma_f32_16x16x4_f32(
                          false, aq[s], false, bk, (short)0, acc, false, false);
            }
            #pragma unroll
            for (int r = 0; r < 8; ++r) {
                const int idx = (r + lg * 8) * CW + tt * 16 + lm;
                acc[r] += fr * cl[idx] + fi * cf[idx];
            }
            S[tt] = acc;
        }

        // ---- online softmax over 64 columns at once -----------------------
        #pragma unroll
        for (int r = 0; r < 8; ++r) {
            float mx = fmaxf(fmaxf(S[0][r], S[1][r]), fmaxf(S[2][r], S[3][r]));
            mx = red16_max(mx);
            float mnew  = fmaxf(mrow[r], mx);
            float alpha = __expf(mrow[r] - mnew);
            mrow[r] = mnew;
            float ps = 0.0f;
            #pragma unroll
            for (int tt = 0; tt < TW; ++tt) {
                float p = __expf(S[tt][r] - mnew);
                S[tt][r] = p;
                ps += p;
            }
            ps = red16_sum(ps);
            lrow[r] = lrow[r] * alpha + ps;
            O[r] *= alpha;
        }

        // ---- P: C-layout -> A-layout via per-wave LDS ---------------------
        #pragma unroll
        for (int tt = 0; tt < TW; ++tt)
            #pragma unroll
            for (int r = 0; r < 8; ++r)
                pw[tt * 256 + (r + lg * 8) * 16 + lm] = S[tt][r];

        // ---- O += P @ V ---------------------------------------------------
        #pragma unroll
        for (int tt = 0; tt < TW; ++tt) {
            const float* vbase = vb0 + c0 + tt * 16 + lg * 2;
            #pragma unroll
            for (int s = 0; s < 4; ++s) {
                v2f ap = *(const v2f*)(pw + tt * 256 + lm * 16 + s * 4 + lg * 2);
                v2f bv = *(const v2f*)(vbase + s * 4);
                O = __builtin_amdgcn_wmma_f32_16x16x4_f32(
                        false, ap, false, bv, (short)0, O, false, false);
            }
        }

        __syncthreads();   // everyone done with tile j before its buffer is reused
    }

    float* cb = ctx + ((size_t)b * N_NODES + n0) * F_DIM + h * HEAD_DIM;
    #pragma unroll
    for (int r = 0; r < 8; ++r)
        cb[(size_t)(r + lg * 8) * F_DIM + lm] = O[r] / lrow[r];
}

// ---------------------------------------------------------------------------
// Kernel 3: out = LN((ctx@Woc+boc)@Woe+boe + residual) ; reg_loss constant
// ---------------------------------------------------------------------------
__global__ __launch_bounds__(256) void out_kernel(
    const float* __restrict__ ctx, const float* __restrict__ feat,
    const float* __restrict__ Woc, const float* __restrict__ boc,
    const float* __restrict__ Woe, const float* __restrict__ boe,
    const float* __restrict__ ln_g, const float* __restrict__ ln_b,
    float* __restrict__ out)
{
    __shared__ float sWoc[F_DIM * C_DIM], sWoe[C_DIM * F_DIM];
    __shared__ float sboc[C_DIM], sboe[F_DIM], sg[F_DIM], sb[F_DIM];
    for (int i = threadIdx.x; i < F_DIM * C_DIM; i += 256) {
        sWoc[i] = Woc[i];
        sWoe[i] = Woe[i];
    }
    if (threadIdx.x < C_DIM) sboc[threadIdx.x] = boc[threadIdx.x];
    if (threadIdx.x < F_DIM) {
        sboe[threadIdx.x] = boe[threadIdx.x];
        sg[threadIdx.x]   = ln_g[threadIdx.x];
        sb[threadIdx.x]   = ln_b[threadIdx.x];
    }
    __syncthreads();

    const int row = blockIdx.x * 256 + threadIdx.x;

    float x[F_DIM];
    #pragma unroll
    for (int d = 0; d < F_DIM; ++d) x[d] = ctx[(size_t)row * F_DIM + d];

    float c8[C_DIM];
    #pragma unroll
    for (int jj = 0; jj < C_DIM; ++jj) {
        float acc = sboc[jj];
        #pragma unroll
        for (int d = 0; d < F_DIM; ++d) acc += x[d] * sWoc[d * C_DIM + jj];
        c8[jj] = acc;
    }

    float y[F_DIM];
    float mu = 0.0f;
    #pragma unroll
    for (int d = 0; d < F_DIM; ++d) {
        float v = sboe[d];
        #pragma unroll
        for (int jj = 0; jj < C_DIM; ++jj) v += c8[jj] * sWoe[jj * F_DIM + d];
        v += feat[(size_t)row * F_DIM + d];
        y[d] = v;
        mu += v;
    }
    mu *= (1.0f / F_DIM);
    float var = 0.0f;
    #pragma unroll
    for (int d = 0; d < F_DIM; ++d) {
        float t = y[d] - mu;
        var += t * t;
    }
    var *= (1.0f / F_DIM);
    const float rs = rsqrtf(var + 1e-5f);
    #pragma unroll
    for (int d = 0; d < F_DIM; ++d)
        out[(size_t)row * F_DIM + d] = (y[d] - mu) * rs * sg[d] + sb[d];

    if (row == 0)
        out[(size_t)BATCH * N_NODES * F_DIM] = 1e-5f / (float)N_NODES;
}

// ---------------------------------------------------------------------------
extern "C" void kernel_launch(void* const* d_in, const int* in_sizes, int n_in,
                              void* d_out, int out_size, void* d_ws, size_t ws_size,
                              hipStream_t stream) {
    const float* feat    = (const float*)d_in[0];
    const float* fixedg  = (const float*)d_in[1];
    const float* learned = (const float*)d_in[2];
    const float* Wqc     = (const float*)d_in[3];
    const float* bqc     = (const float*)d_in[4];
    const float* Wqe     = (const float*)d_in[5];
    const float* bqe     = (const float*)d_in[6];
    const float* Woc     = (const float*)d_in[7];
    const float* boc     = (const float*)d_in[8];
    const float* Woe     = (const float*)d_in[9];
    const float* boe     = (const float*)d_in[10];
    const float* bf1     = (const float*)d_in[11];
    const float* bf2     = (const float*)d_in[12];
    const float* gfusion = (const float*)d_in[13];
    const float* ln_g    = (const float*)d_in[14];
    const float* ln_b    = (const float*)d_in[15];

    const size_t qa_elems = (size_t)BATCH * N_HEADS * N_NODES * KA;
    const size_t vT_elems = (size_t)BATCH * N_HEADS * HEAD_DIM * N_NODES;

    float* qa  = (float*)d_ws;
    float* ka  = qa + qa_elems;
    float* vT  = ka + qa_elems;
    float* ctx = vT + vT_elems;
    (void)ws_size; (void)in_sizes; (void)n_in; (void)out_size;

    prep_kernel<<<(BATCH * N_NODES) / 256, 256, 0, stream>>>(
        feat, Wqc, bqc, Wqe, bqe, bf1, bf2, qa, ka, vT);

    attn_kernel<<<BATCH * (N_NODES / 16), 128, 0, stream>>>(
        qa, ka, vT, learned, fixedg, gfusion, ctx);

    out_kernel<<<(BATCH * N_NODES) / 256, 256, 0, stream>>>(
        ctx, feat, Woc, boc, Woe, boe, ln_g, ln_b, (float*)d_out);
}